// DirectNormLoss_14645838480045
// MI455X (gfx1250) — compile-verified
//
#include <hip/hip_runtime.h>

// ---------------------------------------------------------------------------
// DirectNormLoss for MI455X (gfx1250, wave32).
// Bandwidth-bound (~264 MB @ 23.3 TB/s ~= 11.4 us). <s_i, c_i> computed as the
// diagonal of S_tile x C_tile^T via V_WMMA_F32_16X16X4_F32 on LDS-staged,
// bank-conflict-padded tiles. Staging uses gfx1250 async global->LDS copies
// (ASYNCcnt) when available. Deterministic two-stage reduction.
// ---------------------------------------------------------------------------

typedef float v2f __attribute__((ext_vector_type(2)));
typedef float v8f __attribute__((ext_vector_type(8)));
typedef int   v4i __attribute__((ext_vector_type(4)));

#define DIMD   2048
#define TILE   16          // rows per block (one 16x16 WMMA tile)
#define BLK    256         // 8 wave32s
#define NWAVE  (BLK / 32)
#define CHUNK  512         // columns staged per LDS pass (4 passes over D)
#define CPITCH (CHUNK + 4) // +16B pad: pitch%64==4 -> conflict-free ds_load_b64
#define NCHUNK (DIMD / CHUNK)

#define ATH_AS1 __attribute__((address_space(1)))
#define ATH_AS3 __attribute__((address_space(3)))

#if defined(__has_builtin) && \
    __has_builtin(__builtin_amdgcn_global_load_async_to_lds_b128)
#define ATH_HAS_ASYNC_LDS 1
#else
#define ATH_HAS_ASYNC_LDS 0
#endif

// Per-lane 16B copy global->LDS. On gfx1250 this is the async-to-LDS path
// (no VGPR round trip, tracked by ASYNCcnt); otherwise a plain b128 copy.
// Builtin signature (from hipcc diagnostic): (v4i AS1*, v4i AS3*, Imm, Imm).
__device__ __forceinline__ void cp_glb_to_lds_b128(float* lds_dst,
                                                   const float* g_src) {
#if ATH_HAS_ASYNC_LDS
  __builtin_amdgcn_global_load_async_to_lds_b128(
      (ATH_AS1 v4i*)(g_src), (ATH_AS3 v4i*)(lds_dst), /*offset=*/0,
      /*cpol=*/0);
#else
  *(float4*)lds_dst = *(const float4*)g_src;
#endif
}

// Drain outstanding async global->LDS copies (barriers do NOT imply this).
__device__ __forceinline__ void async_copies_wait_all() {
#if ATH_HAS_ASYNC_LDS
#if defined(__has_builtin) && __has_builtin(__builtin_amdgcn_s_wait_asynccnt)
  __builtin_amdgcn_s_wait_asynccnt(0);
#else
  asm volatile("s_wait_asynccnt 0x0" ::: "memory");
#endif
#endif
}

__device__ __forceinline__ float wave_sum(float v) {
#pragma unroll
  for (int off = 16; off > 0; off >>= 1) v += __shfl_xor(v, off, 32);
  return v;
}

// Kernel 1: inv_cnorm[k] = 1 / ||T_EMB[k]||  (1000 rows, trivial cost)
__global__ void __launch_bounds__(BLK) center_inorm_kernel(
    const float* __restrict__ T, float* __restrict__ icn) {
  __shared__ float red[NWAVE];
  const int row = blockIdx.x;
  const float4* r = (const float4*)(T + (size_t)row * DIMD);
  float s = 0.f;
  for (int i = threadIdx.x; i < DIMD / 4; i += BLK) {
    float4 v = r[i];
    s += v.x * v.x + v.y * v.y + v.z * v.z + v.w * v.w;
  }
  s = wave_sum(s);
  if ((threadIdx.x & 31) == 0) red[threadIdx.x >> 5] = s;
  __syncthreads();
  if (threadIdx.x == 0) {
    float t = 0.f;
#pragma unroll
    for (int w = 0; w < NWAVE; ++w) t += red[w];
    icn[row] = 1.0f / sqrtf(t);
  }
}

// Kernel 2: main tile kernel. One block = 16 rows of the batch.
__global__ void __launch_bounds__(BLK) nd_loss_main_kernel(
    const float* __restrict__ s_emb, const float* __restrict__ t_emb,
    const float* __restrict__ T_EMB, const int* __restrict__ labels,
    const float* __restrict__ icn, float* __restrict__ partial) {
  extern __shared__ float lds[];
  float* lds_s   = lds;                      // TILE * CPITCH
  float* lds_c   = lds + TILE * CPITCH;      // TILE * CPITCH
  float* lds_acc = lds + 2 * TILE * CPITCH;  // NWAVE * 32 * 8

  __shared__ float sh_icn[TILE], sh_sn2[TILE], sh_tn2[TILE], sh_loss[TILE];
  __shared__ int   sh_cls[TILE];

  const int tid  = threadIdx.x;
  const int lane = tid & 31;
  const int wave = tid >> 5;
  const size_t row0 = (size_t)blockIdx.x * TILE;

  if (tid < TILE) {
    int c = labels[row0 + tid];
    sh_cls[tid] = c;
    sh_icn[tid] = icn[c];
  }

  // ||t_i||^2 straight from global (2 rows per wave), overlaps with staging.
  for (int r = wave; r < TILE; r += NWAVE) {
    const float4* tr = (const float4*)(t_emb + (row0 + r) * (size_t)DIMD);
    float s2 = 0.f;
#pragma unroll 4
    for (int i = lane; i < DIMD / 4; i += 32) {
      float4 v = tr[i];
      s2 += v.x * v.x + v.y * v.y + v.z * v.z + v.w * v.w;
    }
    s2 = wave_sum(s2);
    if (lane == 0) sh_tn2[r] = s2;
  }
  __syncthreads();  // sh_cls valid for staging, tn2 stored

  // WMMA operand addressing (see CDNA5 ISA 32-bit A 16x4 / C 16x16 layouts):
  // lane m (0-15): row m, K pair {k,k+1}; lane m+16: row m, K pair {k+2,k+3}.
  // Fold the per-wave K-slice offset into the base pointers so the WMMA loop
  // has a compile-time trip count (no EXEC-mask loop machinery).
  const int m  = lane & 15;
  const int hi = lane >> 4;
  const int kslice = wave * (CHUNK / NWAVE);
  const float* a_ptr = &lds_s[m * CPITCH + 2 * hi + kslice];
  const float* b_ptr = &lds_c[m * CPITCH + 2 * hi + kslice];

  v8f acc = {};
  float sn2_lo = 0.f, sn2_hi = 0.f;  // rows `wave` and `wave+8`

  for (int ch = 0; ch < NCHUNK; ++ch) {
    __syncthreads();  // previous chunk fully consumed before overwrite
    const int colbase4 = ch * (CHUNK / 4);
    // Stage 16 x CHUNK of S and gathered centers into LDS (async b128).
    for (int idx = tid; idx < TILE * (CHUNK / 4); idx += BLK) {
      const int r  = idx >> 7;   // CHUNK/4 == 128
      const int c4 = idx & 127;
      const float* srow = s_emb + (row0 + r) * (size_t)DIMD;
      const float* crow = T_EMB + (size_t)sh_cls[r] * DIMD;
      cp_glb_to_lds_b128(&lds_s[r * CPITCH + c4 * 4],
                         srow + (colbase4 + c4) * 4);
      cp_glb_to_lds_b128(&lds_c[r * CPITCH + c4 * 4],
                         crow + (colbase4 + c4) * 4);
      if (ch + 1 < NCHUNK)  // gfx1250 global_prefetch_b8 for the next chunk
        __builtin_prefetch(
            (const void*)(srow + (colbase4 + (CHUNK / 4) + c4) * 4), 0, 0);
    }
    async_copies_wait_all();  // ASYNCcnt -> 0 before making LDS visible
    __syncthreads();          // staged data visible to all waves

    // ||s_i||^2 from the staged copy (no extra HBM traffic).
    {
      float s2 = 0.f;
#pragma unroll
      for (int i = lane; i < CHUNK / 4; i += 32) {
        float4 v = *(const float4*)&lds_s[wave * CPITCH + i * 4];
        s2 += v.x * v.x + v.y * v.y + v.z * v.z + v.w * v.w;
      }
      sn2_lo += s2;
      s2 = 0.f;
#pragma unroll
      for (int i = lane; i < CHUNK / 4; i += 32) {
        float4 v = *(const float4*)&lds_s[(wave + 8) * CPITCH + i * 4];
        s2 += v.x * v.x + v.y * v.y + v.z * v.z + v.w * v.w;
      }
      sn2_hi += s2;
    }

    // Each wave owns a 64-wide K slice of this chunk: 16 WMMAs per wave.
#pragma unroll 4
    for (int kk = 0; kk < CHUNK / NWAVE; kk += 4) {
      v2f a = *(const v2f*)(a_ptr + kk);
      v2f b = *(const v2f*)(b_ptr + kk);
      acc = __builtin_amdgcn_wmma_f32_16x16x4_f32(
          /*neg_a=*/false, a, /*neg_b=*/false, b,
          /*c_mod=*/(short)0, acc, /*reuse_a=*/false, /*reuse_b=*/false);
    }
  }

  // Finish the per-row sum-of-squares reductions.
  sn2_lo = wave_sum(sn2_lo);
  sn2_hi = wave_sum(sn2_hi);
  if (lane == 0) {
    sh_sn2[wave]     = sn2_lo;
    sh_sn2[wave + 8] = sn2_hi;
  }

  // Spill each wave's 16x16 partial C to LDS, then sum + take the diagonal.
#pragma unroll
  for (int j = 0; j < 8; ++j) lds_acc[wave * 256 + lane * 8 + j] = acc[j];
  __syncthreads();

  if (tid < TILE) {
    // C/D layout: diag(m) lives in lane m, vgpr m (m<8) or lane m+16,
    // vgpr m-8 (m>=8).
    const int l = (tid < 8) ? tid : tid + 16;
    const int j = (tid < 8) ? tid : tid - 8;
    float dot = 0.f;
#pragma unroll
    for (int w = 0; w < NWAVE; ++w) dot += lds_acc[w * 256 + l * 8 + j];
    const float maxn = sqrtf(fmaxf(sh_sn2[tid], sh_tn2[tid]));
    sh_loss[tid] = 1.0f - dot * sh_icn[tid] / maxn;
  }
  __syncthreads();
  if (tid == 0) {
    float s = 0.f;
#pragma unroll
    for (int i = 0; i < TILE; ++i) s += sh_loss[i];
    partial[blockIdx.x] = s;
  }
}

// Kernel 3: deterministic fixed-order final reduction.
__global__ void __launch_bounds__(BLK) nd_loss_reduce_kernel(
    const float* __restrict__ partial, int n, float* __restrict__ out,
    float scale) {
  __shared__ float sh[BLK];
  float s = 0.f;
  for (int i = threadIdx.x; i < n; i += BLK) s += partial[i];
  sh[threadIdx.x] = s;
  __syncthreads();
#pragma unroll
  for (int off = BLK / 2; off > 0; off >>= 1) {
    if (threadIdx.x < off) sh[threadIdx.x] += sh[threadIdx.x + off];
    __syncthreads();
  }
  if (threadIdx.x == 0) out[0] = sh[0] * scale;
}

extern "C" void kernel_launch(void* const* d_in, const int* in_sizes, int n_in,
                              void* d_out, int out_size, void* d_ws,
                              size_t ws_size, hipStream_t stream) {
  const float* s_emb  = (const float*)d_in[0];
  const float* t_emb  = (const float*)d_in[1];
  const float* T_EMB  = (const float*)d_in[2];
  const int*   labels = (const int*)d_in[3];

  const int B   = in_sizes[3];            // 16384
  const int NCL = in_sizes[2] / DIMD;     // 1000
  const int nblocks = B / TILE;           // 1024

  float* icn     = (float*)d_ws;          // NCL floats (padded to 1024)
  float* partial = icn + 1024;            // nblocks floats

  center_inorm_kernel<<<NCL, BLK, 0, stream>>>(T_EMB, icn);

  const size_t shmem =
      (size_t)(2 * TILE * CPITCH + NWAVE * 32 * 8) * sizeof(float);
  (void)hipFuncSetAttribute(reinterpret_cast<const void*>(nd_loss_main_kernel),
                            hipFuncAttributeMaxDynamicSharedMemorySize,
                            (int)shmem);
  nd_loss_main_kernel<<<nblocks, BLK, shmem, stream>>>(
      s_emb, t_emb, T_EMB, labels, icn, partial);

  nd_loss_reduce_kernel<<<1, BLK, 0, stream>>>(partial, nblocks, (float*)d_out,
                                               1.0f / (float)B);
}